// Linear_11381663334678
// MI455X (gfx1250) — compile-verified
//
#include <hip/hip_runtime.h>
#include <hip/hip_bf16.h>

// MI455X / gfx1250 — spherical-harmonic block-diagonal linear (e3nn-style)
// y[n,m,:] = x[n,m,:] @ W[l(m)] * (1/sqrt(128))
// Strategy: exact-fp32 path using V_WMMA_F32_16X16X4_F32 (no precision loss,
// no fp32->bf16 conversion VALU). One wave computes a 16-node x 128-cout slab
// for a fixed m; K=128 consumed in 32 chunks of 4.

typedef __attribute__((ext_vector_type(2))) float v2f;
typedef __attribute__((ext_vector_type(8))) float v8f;

#define N_M     16      // (LMAX+1)^2 spherical components
#define C_IN    128
#define C_OUT   128
#define PATH_W  0.08838834764831845f   // 1/sqrt(128)

// ---------------------------------------------------------------------------
// Prep: scatter W[l][k][c] * PATH_W into the exact WMMA-B per-lane layout.
// Out layout (floats): [l(4)][ct(8)][kc(32)][lane(32)][vj(2)]
//   lane's element for VGPR vj:  k = kc*4 + vj + 2*(lane>>4), c = ct*16 + (lane&15)
// (ISA 32-bit B 4x16 striping: VGPR0 holds K=0 in lanes 0-15, K=2 in lanes
//  16-31; VGPR1 holds K=1 / K=3; lane%16 = output column.)
// Total: 4*8*32*32*2 = 65536 floats = 256 KB in d_ws.
// ---------------------------------------------------------------------------
__global__ __launch_bounds__(256) void prep_weights_k(const float* __restrict__ w,
                                                      float* __restrict__ bbuf) {
    int t = blockIdx.x * 256 + threadIdx.x;       // 0..65535
    int vj   = t & 1;
    int lane = (t >> 1) & 31;
    int kc   = (t >> 6) & 31;
    int ct   = (t >> 11) & 7;
    int l    = (t >> 14) & 3;
    int k = kc * 4 + vj + 2 * (lane >> 4);
    int c = ct * 16 + (lane & 15);
    bbuf[t] = PATH_W * w[((size_t)(l * C_IN + k)) * C_OUT + c];
}

// ---------------------------------------------------------------------------
// GEMM: one wave handles (node-block nb of 16 nodes, one m).
//   A-tile: rows = 16 nodes (row stride 16*128 floats), K = 128.
//   A frag for K-chunk kc: lane (row = lane&15, h = lane>>4) loads the
//   contiguous pair x[.., kc*4 + 2h .. kc*4 + 2h + 1]  -> one b64 load.
//   Entire A held in 32 float2 (64 VGPRs), reused across all 8 col tiles.
//   Per col tile: 32 x v_wmma_f32_16x16x4_f32 accumulation chain.
// ---------------------------------------------------------------------------
__global__ __launch_bounds__(256) void sh_linear_wmma_k(const float* __restrict__ x,
                                                        const float* __restrict__ bbuf,
                                                        float* __restrict__ y) {
    const int tid  = threadIdx.x;
    const int lane = tid & 31;
    const int gw   = blockIdx.x * 8 + (tid >> 5);   // global wave id, 0..N_nodes-1
    const int nb   = gw >> 4;                       // node block (16 nodes)
    const int m    = gw & 15;                       // spherical component
    // l = floor(sqrt(m)) for m in [0,16)
    const int l = (m >= 9) ? 3 : (m >= 4) ? 2 : (m >= 1) ? 1 : 0;

    const int row = lane & 15;                      // node within block / out col
    const int h   = lane >> 4;                      // K-half select

    // ---- load full A tile: 32 chunks of K=4, one b64 per lane per chunk ----
    const float* xa = x + ((size_t)(nb * 16 + row) * N_M + m) * C_IN + 2 * h;
    v2f a[32];
#pragma unroll
    for (int kc = 0; kc < 32; ++kc) {
        a[kc] = *(const v2f*)(xa + kc * 4);
    }

    const float* bl = bbuf + (size_t)l * (8 * 32 * 64);          // this l's B block
    float* yo = y + ((size_t)(nb * 16) * N_M + m) * C_OUT;        // node base, this m

#pragma unroll
    for (int ct = 0; ct < 8; ++ct) {
        // coalesced b64 B-frag loads: [kc][lane][2]
        const v2f* bp = (const v2f*)(bl + (size_t)ct * (32 * 64)) + lane;
        v8f acc = {};
#pragma unroll
        for (int kc = 0; kc < 32; ++kc) {
            v2f b = bp[kc * 32];
            // (neg_a, A, neg_b, B, c_mod, C, reuse_a, reuse_b)
            acc = __builtin_amdgcn_wmma_f32_16x16x4_f32(
                false, a[kc], false, b, (short)0, acc, false, false);
        }
        // D layout: VGPR r, lane -> node row (r + 8h), col (ct*16 + lane&15)
#pragma unroll
        for (int r = 0; r < 8; ++r) {
            yo[(size_t)(r + 8 * h) * (N_M * C_OUT) + ct * 16 + row] = acc[r];
        }
    }
}

extern "C" void kernel_launch(void* const* d_in, const int* in_sizes, int n_in,
                              void* d_out, int out_size, void* d_ws, size_t ws_size,
                              hipStream_t stream) {
    const float* x = (const float*)d_in[0];   // [N, 16, 128] fp32
    const float* w = (const float*)d_in[1];   // [4, 128, 128] fp32
    float* y    = (float*)d_out;              // [N, 16, 128] fp32
    float* bbuf = (float*)d_ws;               // 256 KB rearranged weights

    // rearrange + scale weights into WMMA-B lane layout (65536 elements)
    prep_weights_k<<<256, 256, 0, stream>>>(w, bbuf);

    const int n_nodes = in_sizes[0] / (N_M * C_IN);   // 100000
    // one wave per (16-node block, m): n_nodes/16 * 16 = n_nodes waves
    const int blocks = n_nodes / 8;                   // 8 waves per 256-thread block
    sh_linear_wmma_k<<<blocks, 256, 0, stream>>>(x, bbuf, y);
}